// QuantumNNLayer_61400852463722
// MI455X (gfx1250) — compile-verified
//
#include <hip/hip_runtime.h>
#include <hip/hip_bf16.h>

typedef float v2f __attribute__((ext_vector_type(2)));
typedef float v8f __attribute__((ext_vector_type(8)));

// lane p <-> lane p+16 swap via ds_swizzle immediate (group-of-32, xor=0x10, and=0x1f)
__device__ __forceinline__ float swap16(float v) {
    return __int_as_float(__builtin_amdgcn_ds_swizzle(__float_as_int(v), 0x401F));
}

// ---------------- complex helpers ----------------
struct c32 { float x, y; };
__device__ __forceinline__ c32 cmul(c32 a, c32 b) {
    return { a.x * b.x - a.y * b.y, a.x * b.y + a.y * b.x };
}

// g = rz(tz) @ ry(ty) @ H2   (2x2 complex)
__device__ __forceinline__ void make_gate(float ty, float tz, c32 g[2][2]) {
    const float is2 = 0.70710678118654752f;
    float cy = __cosf(0.5f * ty), sy = __sinf(0.5f * ty);
    float cz = __cosf(0.5f * tz), sz = __sinf(0.5f * tz);
    c32 em = { cz, -sz };   // e^{-i tz/2}
    c32 ep = { cz,  sz };   // e^{+i tz/2}
    g[0][0] = { em.x * (cy - sy) * is2, em.y * (cy - sy) * is2 };
    g[0][1] = { em.x * (cy + sy) * is2, em.y * (cy + sy) * is2 };
    g[1][0] = { ep.x * (sy + cy) * is2, ep.y * (sy + cy) * is2 };
    g[1][1] = { ep.x * (sy - cy) * is2, ep.y * (sy - cy) * is2 };
}

// GH[s][col] for GH = kron(a,b,a,b); wire0 = MSB
__device__ __forceinline__ c32 gh_elem(const c32 a[2][2], const c32 b[2][2], int s, int col) {
    c32 r = cmul(a[(s >> 3) & 1][(col >> 3) & 1], b[(s >> 2) & 1][(col >> 2) & 1]);
    r = cmul(r, a[(s >> 1) & 1][(col >> 1) & 1]);
    r = cmul(r, b[s & 1][col & 1]);
    return r;
}

// ---------------- setup: M[j][k] = Re( sum_s c(perm(s)) GH[s][j] conj(GH[s][k]) ) ----------
__global__ void qnn_build_M(const float* __restrict__ w0, const float* __restrict__ w1,
                            float* __restrict__ M) {
    int t = threadIdx.x;
    if (t >= 256) return;
    int j = t >> 4, k = t & 15;

    c32 ga[2][2], gb[2][2];
    make_gate(w0[0], w1[0], ga);
    make_gate(w0[1], w1[1], gb);

    float acc = 0.0f;
    for (int s = 0; s < 16; ++s) {
        // CNOT chain permutation (wire0 = MSB): (0,1),(1,2),(2,3),(3,0)
        int b0 = (s >> 3) & 1, b1 = (s >> 2) & 1, b2 = (s >> 1) & 1, b3 = s & 1;
        b1 ^= b0; b2 ^= b1; b3 ^= b2; b0 ^= b3;
        int i2 = (b0 << 3) | (b1 << 2) | (b2 << 1) | b3;
        float ci = 1.0f - 0.25f * (float)__popc(i2);
        c32 gj = gh_elem(ga, gb, s, j);
        c32 gk = gh_elem(ga, gb, s, k);
        acc += ci * (gj.x * gk.x + gj.y * gk.y);   // Re(gj * conj(gk))
    }
    M[j * 16 + k] = acc;
}

// ---------------- main kernel: 16 pixels per wave, WMMA f32 16x16x4 chain -----------------
// lane p (0..15) holds patch row fh=0 of pixel p, lane p+16 holds row fh=1.
__global__ void __launch_bounds__(256)
qnn_main(const float* __restrict__ x, const float* __restrict__ M, float* __restrict__ out) {
    const int lane = threadIdx.x & 31;
    const int wid  = (blockIdx.x * blockDim.x + threadIdx.x) >> 5;
    const int p  = lane & 15;
    const int hi = lane >> 4;
    const int q  = wid * 16 + p;            // global pixel id (grid sized exactly)

    const int OW = 258, PER_IMG = 258 * 258;
    int b   = q / PER_IMG;
    int rem = q - b * PER_IMG;
    int i   = rem / OW;
    int j   = rem - i * OW;

    // padded patch: row i-1+hi, cols j-1, j  (pad(1,2) on H and W)
    int r = i + hi - 1;
    bool okr = (unsigned)r < 256u;
    bool ok0 = okr && ((unsigned)(j - 1) < 256u);
    bool ok1 = okr && ((unsigned)j < 256u);

    // clamp addresses so every load is valid; zero-mask with cndmask (no divergent branches)
    int rc = min(255, max(0, r));
    int c0 = min(255, max(0, j - 1));
    int c1 = min(255, j);

    v2f rv[4];
    #pragma unroll
    for (int c = 0; c < 4; ++c) {
        int base = ((b * 4 + c) * 256 + rc) * 256;
        float a0 = x[base + c0];
        float a1 = x[base + c1];
        rv[c].x = ok0 ? a0 : 0.0f;
        rv[c].y = ok1 ? a1 : 0.0f;
    }

    // A operand (M matrix), 16x4 f32 chunks:
    //   VGPR0: K=4c (lanes 0-15) / K=4c+2 (lanes 16-31); VGPR1: K=4c+1 / K=4c+3
    const float* Mr = M + p * 16 + hi * 2;
    v2f A0 = { Mr[0],  Mr[1]  };
    v2f A1 = { Mr[4],  Mr[5]  };
    v2f A2 = { Mr[8],  Mr[9]  };
    v2f A3 = { Mr[12], Mr[13] };

    // B operand chunk c is exactly (rv[c].x, rv[c].y) per lane — no shuffles needed.
    v8f acc = {};
    acc = __builtin_amdgcn_wmma_f32_16x16x4_f32(false, A0, false, rv[0], (short)0, acc, false, false);
    acc = __builtin_amdgcn_wmma_f32_16x16x4_f32(false, A1, false, rv[1], (short)0, acc, false, false);
    acc = __builtin_amdgcn_wmma_f32_16x16x4_f32(false, A2, false, rv[2], (short)0, acc, false, false);
    acc = __builtin_amdgcn_wmma_f32_16x16x4_f32(false, A3, false, rv[3], (short)0, acc, false, false);

    // Each lane dots its LOCAL u half (acc[0..7] = u[hi*8 .. hi*8+7]) with the matching v half.
    // Missing v values live in the partner lane; pre-select so one swizzle set serves both halves:
    //   lower lane sends ch2/ch3 row-0 (f8,f9,f12,f13), upper lane sends ch0/ch1 row-1 (f2,f3,f6,f7)
    float s0 = swap16(hi ? rv[0].x : rv[2].x);
    float s1 = swap16(hi ? rv[0].y : rv[2].y);
    float s2 = swap16(hi ? rv[1].x : rv[3].x);
    float s3 = swap16(hi ? rv[1].y : rv[3].y);

    float m0 = hi ? s0      : rv[0].x;
    float m1 = hi ? s1      : rv[0].y;
    float m2 = hi ? rv[2].x : s0;
    float m3 = hi ? rv[2].y : s1;
    float m4 = hi ? s2      : rv[1].x;
    float m5 = hi ? s3      : rv[1].y;
    float m6 = hi ? rv[3].x : s2;
    float m7 = hi ? rv[3].y : s3;

    float num = m0 * acc[0] + m1 * acc[1] + m2 * acc[2] + m3 * acc[3]
              + m4 * acc[4] + m5 * acc[5] + m6 * acc[6] + m7 * acc[7];
    float den = rv[0].x * rv[0].x + rv[0].y * rv[0].y + rv[1].x * rv[1].x + rv[1].y * rv[1].y
              + rv[2].x * rv[2].x + rv[2].y * rv[2].y + rv[3].x * rv[3].x + rv[3].y * rv[3].y;

    num += swap16(num);
    den += swap16(den);

    if (hi == 0) out[q] = num / den;   // vᵀMv / vᵀv  (0/0 -> NaN at all-pad corners, as reference)
}

extern "C" void kernel_launch(void* const* d_in, const int* in_sizes, int n_in,
                              void* d_out, int out_size, void* d_ws, size_t ws_size,
                              hipStream_t stream) {
    const float* x  = (const float*)d_in[0];
    const float* w0 = (const float*)d_in[1];
    const float* w1 = (const float*)d_in[2];
    float* out = (float*)d_out;
    float* M   = (float*)d_ws;          // 256 floats

    qnn_build_M<<<1, 256, 0, stream>>>(w0, w1, M);

    const int N      = 32 * 258 * 258;  // 2,130,048 pixels
    const int waves  = N / 16;          // 133,128 (exact)
    const int blocks = waves / 8;       // 256 threads = 8 waves per block (exact)
    qnn_main<<<blocks, 256, 0, stream>>>(x, M, out);
}